// SingleStageDetector_40407052321284
// MI455X (gfx1250) — compile-verified
//
#include <hip/hip_runtime.h>
#include <hip/hip_bf16.h>

typedef __attribute__((ext_vector_type(2))) float v2f;
typedef __attribute__((ext_vector_type(8))) float v8f;

// Problem constants (from reference)
constexpr int kB    = 512;   // batch
constexpr int kCIN  = 1280;  // input channels
constexpr int kHID  = 128;   // hidden
constexpr int kCOUT = 65;    // 5*A + C
constexpr int kA    = 9;     // anchors
constexpr int kC    = 20;    // classes
constexpr int kS    = 49;    // 7*7 spatial
constexpr int kSP   = 64;    // padded spatial
constexpr int kM    = 2048;  // sampled anchors
constexpr int kNBOX = 40;    // gt boxes per image
constexpr int kW2P  = 80;    // padded rows of w2 (5 tiles of 16)

__constant__ float cANC[kA][2] = {
    {1.f,1.f},{2.f,2.f},{3.f,3.f},{4.f,4.f},{5.f,5.f},
    {2.f,3.f},{3.f,2.f},{3.f,5.f},{5.f,3.f}};

__device__ __forceinline__ float sigmoidf(float x) {
    return 1.f / (1.f + __expf(-x));
}

// ---------------------------------------------------------------------------
// Prep: w1 -> w1t (1280x128, plain transpose),
//       w2 -> w2t (K-pair-interleaved 128x80: elem(k,m) at (k>>1)*160+m*2+(k&1),
//                  zero-padded rows m>=65),
//       zero the loss accumulator.
// ---------------------------------------------------------------------------
__global__ void prep_kernel(const float* __restrict__ w1,
                            const float* __restrict__ w2,
                            float* __restrict__ w1t,
                            float* __restrict__ w2t,
                            float* __restrict__ out_loss) {
    int i = blockIdx.x * blockDim.x + threadIdx.x;
    if (i == 0) out_loss[0] = 0.f;
    if (i < kCIN * kHID) {
        int k = i / kHID, m = i % kHID;
        w1t[i] = w1[m * kCIN + k];
    }
    int j = i - kCIN * kHID;
    if (j >= 0 && j < kHID * kW2P) {
        int k = j / kW2P, m = j % kW2P;
        float v = (m < kCOUT) ? w2[m * kHID + k] : 0.f;
        w2t[(k >> 1) * (2 * kW2P) + m * 2 + (k & 1)] = v;
    }
}

// ---------------------------------------------------------------------------
// Fused MLP head: per image b,
//   H1 = leaky(w1 @ F_b + b1)   (128 x 49)   [f32 WMMA 16x16x4]
//   OUT = w2 @ H1 + b2          (65  x 49)
// One block per image, 8 wave32s. All LDS operands are K-pair interleaved so
// each v2f WMMA fragment is one aligned b64 load.
// ---------------------------------------------------------------------------
__global__ __launch_bounds__(256) void fused_head_kernel(
    const float* __restrict__ feats,   // (B, 1280, 7, 7)
    const float* __restrict__ w1t,     // (1280, 128) plain
    const float* __restrict__ b1,      // (128)
    const float* __restrict__ w2t,     // (128, 80) K-pair interleaved
    const float* __restrict__ b2,      // (65)
    float* __restrict__ outbuf)        // (B, 65, 49)
{
    // interleaved layouts: elem(k,c) at (k>>1)*2W + c*2 + (k&1)
    __shared__ __align__(16) float ldsH[kHID * kSP];   // 32 KB: H1 128x64
    __shared__ __align__(16) float ldsF[32 * kSP];     //  8 KB: F chunk 32x64
    __shared__ __align__(16) float ldsW[32 * kHID];    // 16 KB: w1t chunk 32x128

    v2f* lds2H = (v2f*)ldsH;   // row-pair stride 64
    v2f* lds2F = (v2f*)ldsF;   // row-pair stride 64
    v2f* lds2W = (v2f*)ldsW;   // row-pair stride 128
    const v2f* w2t2 = (const v2f*)w2t;  // row-pair stride 80

    const int b    = blockIdx.x;
    const int tid  = threadIdx.x;
    const int lane = tid & 31;
    const int wave = tid >> 5;        // 0..7 == M-tile in phase 1
    const int half = lane >> 4;       // 0 or 1
    const int l16  = lane & 15;

    const float* F = feats + (size_t)b * kCIN * kS;

    v8f acc[4];
    #pragma unroll
    for (int nt = 0; nt < 4; ++nt) acc[nt] = (v8f){0.f,0.f,0.f,0.f,0.f,0.f,0.f,0.f};

    const int mrow = wave * 16 + l16;

    for (int k0 = 0; k0 < kCIN; k0 += 32) {
        // Stage feature chunk (32 x 49 -> 32 x 64, zero padded, interleaved)
        for (int s = tid; s < 32 * kSP; s += 256) {
            int r = s >> 6, c = s & 63;
            float v = (c < kS) ? F[(k0 + r) * kS + c] : 0.f;
            ldsF[(r >> 1) * (2 * kSP) + c * 2 + (r & 1)] = v;
        }
        // Stage w1t chunk (32 x 128, interleaved)
        for (int s = tid; s < 32 * kHID; s += 256) {
            int rk = s >> 7, m = s & 127;
            ldsW[(rk >> 1) * (2 * kHID) + m * 2 + (rk & 1)] = w1t[(size_t)k0 * kHID + s];
        }
        if (k0 + 32 < kCIN) {
            __builtin_prefetch(F + (size_t)(k0 + 32) * kS, 0, 0);
        }
        __syncthreads();

        #pragma unroll
        for (int k = 0; k < 32; k += 4) {
            const int kb2 = (k >> 1) + half;   // even pair base + this half's pair
            v2f a = lds2W[kb2 * kHID + mrow];
            #pragma unroll
            for (int nt = 0; nt < 4; ++nt) {
                v2f bb = lds2F[kb2 * kSP + nt * 16 + l16];
                acc[nt] = __builtin_amdgcn_wmma_f32_16x16x4_f32(
                    false, a, false, bb, (short)0, acc[nt], false, false);
            }
        }
        __syncthreads();
    }

    // H1 = leaky(acc + b1) -> LDS (interleaved over hidden index)
    #pragma unroll
    for (int nt = 0; nt < 4; ++nt) {
        #pragma unroll
        for (int r = 0; r < 8; ++r) {
            int row = wave * 16 + half * 8 + r;
            int col = nt * 16 + l16;
            float v = acc[nt][r] + b1[row];
            v = (v > 0.f) ? v : 0.01f * v;
            ldsH[(row >> 1) * (2 * kSP) + col * 2 + (row & 1)] = v;
        }
    }
    __syncthreads();

    // Phase 2: OUT = w2 @ H1 + b2 ; 5 M-tiles x 4 N-tiles over 8 waves
    for (int t = wave; t < 20; t += 8) {
        int mt = t / 4, nt = t % 4;
        v8f c2 = (v8f){0.f,0.f,0.f,0.f,0.f,0.f,0.f,0.f};
        #pragma unroll 8
        for (int k = 0; k < kHID; k += 4) {
            const int kb2 = (k >> 1) + half;
            v2f a  = w2t2[kb2 * kW2P + mt * 16 + l16];
            v2f bb = lds2H[kb2 * kSP + nt * 16 + l16];
            c2 = __builtin_amdgcn_wmma_f32_16x16x4_f32(
                false, a, false, bb, (short)0, c2, false, false);
        }
        #pragma unroll
        for (int r = 0; r < 8; ++r) {
            int row = mt * 16 + half * 8 + r;
            int col = nt * 16 + l16;
            if (row < kCOUT && col < kS) {
                outbuf[((size_t)b * kCOUT + row) * kS + col] = c2[r] + b2[row];
            }
        }
    }
}

// ---------------------------------------------------------------------------
// Losses over sampled anchors. One thread per sample (M=2048), block-reduce,
// atomicAdd into d_out[0].
// ---------------------------------------------------------------------------
__global__ __launch_bounds__(256) void loss_kernel(
    const float* __restrict__ outbuf,     // (B, 65, 49)
    const int*  __restrict__ pos_idx,     // (M)
    const int*  __restrict__ neg_idx,     // (M)
    const float* __restrict__ gt_off,     // (M, 4)
    const int*  __restrict__ gt_cls,      // (M)
    float* __restrict__ out_loss)
{
    __shared__ float red[256];
    const int m = blockIdx.x * blockDim.x + threadIdx.x;
    float part = 0.f;
    if (m < kM) {
        const float invM = 1.f / (float)kM;
        // positive sample
        int pi = pos_idx[m];
        int pb = pi / (kA * kS);
        int pr = pi % (kA * kS);
        int pa = pr / kS;
        int ps = pr % kS;
        const float* ob = outbuf + (size_t)pb * kCOUT * kS;
        float cp = sigmoidf(ob[(pa * 5 + 0) * kS + ps]);
        part += (cp - 1.f) * (cp - 1.f) * (0.5f * invM);
        // negative sample
        int ni = neg_idx[m];
        int nb = ni / (kA * kS);
        int nr = ni % (kA * kS);
        int na = nr / kS;
        int ns = nr % kS;
        const float* obn = outbuf + (size_t)nb * kCOUT * kS;
        float cn = sigmoidf(obn[(na * 5 + 0) * kS + ns]);
        part += cn * cn * (0.5f * invM);
        // regression loss (channels a*5+1..a*5+4; first two sigmoid-0.5)
        #pragma unroll
        for (int j = 1; j <= 4; ++j) {
            float raw = ob[(pa * 5 + j) * kS + ps];
            float t = (j <= 2) ? (sigmoidf(raw) - 0.5f) : raw;
            float d = t - gt_off[m * 4 + (j - 1)];
            part += d * d * invM;
        }
        // class loss: log-softmax over 20 class channels (45..64)
        float xs[kC];
        float mx = -1e30f;
        #pragma unroll
        for (int c = 0; c < kC; ++c) {
            xs[c] = ob[(5 * kA + c) * kS + ps];
            mx = fmaxf(mx, xs[c]);
        }
        float se = 0.f;
        #pragma unroll
        for (int c = 0; c < kC; ++c) se += __expf(xs[c] - mx);
        float lp = xs[gt_cls[m]] - mx - __logf(se);
        part += -lp / (float)kB;
    }
    red[threadIdx.x] = part;
    __syncthreads();
    for (int o = 128; o > 0; o >>= 1) {
        if (threadIdx.x < o) red[threadIdx.x] += red[threadIdx.x + o];
        __syncthreads();
    }
    if (threadIdx.x == 0) atomicAdd(out_loss, red[0]);
}

// ---------------------------------------------------------------------------
// IoU matrix: (B, 441, 40), replicating reference formula exactly.
// ---------------------------------------------------------------------------
__global__ __launch_bounds__(256) void iou_kernel(
    const float* __restrict__ bboxes,  // (B, 40, 5)
    float* __restrict__ iou,           // (B, 441, 40) flat
    long long total)
{
    long long i = (long long)blockIdx.x * blockDim.x + threadIdx.x;
    if (i >= total) return;
    int n = (int)(i % kNBOX);
    long long r = i / kNBOX;
    int p = (int)(r % (kA * kS));
    int b = (int)(r / (kA * kS));
    int a = p / kS;
    int s = p % kS;
    int h = s / 7;
    int w = s % 7;
    float cx = w + 0.5f, cy = h + 0.5f;
    float hw = cANC[a][0] * 0.5f, hh2 = cANC[a][1] * 0.5f;
    float x = cx - hw, X = cx + hw;
    float y = cy - hh2, Y = cy + hh2;
    const float* bb = bboxes + ((size_t)b * kNBOX + n) * 5;
    float xb = bb[0], yb = bb[1], Xb = bb[2], Yb = bb[3];
    float e1 = X - xb, e2 = Xb - x;
    float f1 = Y - yb, f2 = Yb - y;
    float ww = fminf(e1, e2);
    float hh = fminf(f1, f2);
    float Wd = fmaxf(e1, e2);
    float Hd = fmaxf(f1, f2);
    if (e1 < 0.f || e2 < 0.f) ww = 0.f;
    iou[i] = ww * hh / (Wd * Hd);
}

// ---------------------------------------------------------------------------
extern "C" void kernel_launch(void* const* d_in, const int* in_sizes, int n_in,
                              void* d_out, int out_size, void* d_ws, size_t ws_size,
                              hipStream_t stream) {
    const float* feats   = (const float*)d_in[0];
    const float* bboxes  = (const float*)d_in[1];
    const int*   pos_idx = (const int*)d_in[2];
    const int*   neg_idx = (const int*)d_in[3];
    const float* gt_off  = (const float*)d_in[4];
    const int*   gt_cls  = (const int*)d_in[5];
    const float* w1      = (const float*)d_in[6];
    const float* b1      = (const float*)d_in[7];
    const float* w2      = (const float*)d_in[8];
    const float* b2      = (const float*)d_in[9];
    float* out = (float*)d_out;

    // Workspace layout (floats)
    float* w1t    = (float*)d_ws;                 // 1280*128
    float* w2t    = w1t + kCIN * kHID;            // 128*80 (interleaved)
    float* outbuf = w2t + kHID * kW2P;            // 512*65*49

    int prepN = kCIN * kHID + kHID * kW2P;
    prep_kernel<<<(prepN + 255) / 256, 256, 0, stream>>>(w1, w2, w1t, w2t, out);

    fused_head_kernel<<<kB, 256, 0, stream>>>(feats, w1t, b1, w2t, b2, outbuf);

    loss_kernel<<<kM / 256, 256, 0, stream>>>(outbuf, pos_idx, neg_idx,
                                              gt_off, gt_cls, out);

    long long total = (long long)kB * (kA * kS) * kNBOX;
    iou_kernel<<<(int)((total + 255) / 256), 256, 0, stream>>>(bboxes, out + 1, total);
}